// SActv_67250597921093
// MI455X (gfx1250) — compile-verified
//
#include <hip/hip_runtime.h>
#include <stdint.h>

typedef uint32_t u32;
typedef int v4i __attribute__((ext_vector_type(4)));

#define WORDS_PER_ROW 256          // 256 x int32 = 8192-bit stream per row
#define ROWS_PER_BLOCK 4
#define BLOCK_THREADS 256
#define WORDS_PER_BLOCK (WORDS_PER_ROW * ROWS_PER_BLOCK)   // 1024 words = 4KB

__device__ __forceinline__ u32 maj3(u32 a, u32 b, u32 c) {
    // (a&b) | (a&c) | (b&c)  ==  (a&b) | (c & (a|b))
    return (a & b) | (c & (a | b));
}

// Circular right-shift of the MSB-first packed bitstream by s (0..8) bits,
// evaluated for one word: needs current word and the row-circular previous word.
// With constant s this folds to a single v_alignbit_b32 (funnel shift).
__device__ __forceinline__ u32 shifted_word(u32 prev, u32 cur, int s) {
    return s ? ((cur >> s) | (prev << (32 - s))) : cur;
}

__global__ __launch_bounds__(BLOCK_THREADS) void sactv_maj9_kernel(
        const u32* __restrict__ in, u32* __restrict__ out) {
    __shared__ __align__(16) u32 lds[WORDS_PER_BLOCK];

    const int t = threadIdx.x;
    const size_t blk_base = (size_t)blockIdx.x * WORDS_PER_BLOCK;
    const u32* blk_in  = in  + blk_base;
    u32*       blk_out = out + blk_base;

    // ---- Stage 4KB (4 rows) of packed stream into LDS -------------------
#if defined(__gfx1250__) && __has_builtin(__builtin_amdgcn_global_load_async_to_lds_b128)
    {
        // CDNA5 async data mover: each lane DMAs 16B global -> LDS, tracked by ASYNCcnt.
        typedef __attribute__((address_space(1))) v4i* gptr_v4i;
        typedef __attribute__((address_space(3))) v4i* lptr_v4i;
        gptr_v4i gsrc = (gptr_v4i)(blk_in + t * 4);
        lptr_v4i ldst = (lptr_v4i)(&lds[t * 4]);
        __builtin_amdgcn_global_load_async_to_lds_b128(gsrc, ldst, 0, 0);
#if __has_builtin(__builtin_amdgcn_s_wait_asynccnt)
        __builtin_amdgcn_s_wait_asynccnt(0);
#else
        asm volatile("s_wait_asynccnt 0" ::: "memory");
#endif
    }
#else
    {
        // Fallback: coalesced 128-bit load + LDS store.
        const uint4 v = *(const uint4*)(blk_in + t * 4);
        *(uint4*)(&lds[t * 4]) = v;
    }
#endif
    __syncthreads();

    // ---- Compute 4 output words per thread ------------------------------
    // Thread t handles words 4t..4t+3 of the block; row r = t/64.
    const int r  = t >> 6;
    const int l0 = (t & 63) << 2;               // row-local word index of first word
    const u32* row = &lds[r * WORDS_PER_ROW];

    u32 res[4];
#pragma unroll
    for (int k = 0; k < 4; ++k) {
        const int l = l0 + k;
        const u32 cur  = row[l];
        const u32 prev = row[(l - 1) & (WORDS_PER_ROW - 1)];   // row-circular wrap

        // 9 shifted streams (s = 0..8), MAJ3 tree: (0,1,2),(3,4,5),(6,7,8) -> maj3
        const u32 m0 = maj3(shifted_word(prev, cur, 0),
                            shifted_word(prev, cur, 1),
                            shifted_word(prev, cur, 2));
        const u32 m1 = maj3(shifted_word(prev, cur, 3),
                            shifted_word(prev, cur, 4),
                            shifted_word(prev, cur, 5));
        const u32 m2 = maj3(shifted_word(prev, cur, 6),
                            shifted_word(prev, cur, 7),
                            shifted_word(prev, cur, 8));
        res[k] = maj3(m0, m1, m2);
    }

    *(uint4*)(blk_out + t * 4) = make_uint4(res[0], res[1], res[2], res[3]);
}

extern "C" void kernel_launch(void* const* d_in, const int* in_sizes, int n_in,
                              void* d_out, int out_size, void* d_ws, size_t ws_size,
                              hipStream_t stream) {
    (void)n_in; (void)d_ws; (void)ws_size; (void)out_size;
    const u32* in = (const u32*)d_in[0];   // int32 packed bitstreams, (2048, 256)
    u32* out = (u32*)d_out;                // int32 output, same shape (raw 32-bit words)

    const int total_words = in_sizes[0];               // 2048 * 256 = 524288
    const int blocks = total_words / WORDS_PER_BLOCK;  // 512
    sactv_maj9_kernel<<<blocks, BLOCK_THREADS, 0, stream>>>(in, out);
}